// AFM_89627377532987
// MI455X (gfx1250) — compile-verified
//
#include <hip/hip_runtime.h>
#include <hip/hip_bf16.h>
#include <math.h>

// Problem constants (from reference)
#define BATCH 4096
#define NF    26          // sparse features
#define VOC   100000
#define DD    16          // embedding dim
#define AA    16          // attention dim
#define NDNS  13          // dense features
#define NPAIR 325         // 26*25/2
#define NTILE 21          // ceil(325/16)
#define ROWS  8           // batch rows per block (one wave each)

typedef float v2f __attribute__((ext_vector_type(2)));
typedef float v8f __attribute__((ext_vector_type(8)));

#if defined(__gfx1250__) && __has_builtin(__builtin_amdgcn_global_load_async_to_lds_b128)
#define HAVE_ASYNC_LDS 1
// 16-byte int vector, matching the builtin's pointee type exactly
typedef int v4i_t __attribute__((vector_size(16)));
typedef v4i_t __attribute__((address_space(1))) v4i_global_t;   // __device__ (global)
typedef v4i_t __attribute__((address_space(3))) v4i_lds_t;      // __shared__ (LDS)
#else
#define HAVE_ASYNC_LDS 0
#endif

__global__ __launch_bounds__(ROWS * 32) void afm_kernel(
    const float* __restrict__ dense_input,   // (B, ND)
    const int*   __restrict__ sparse_input,  // (B, F)
    const float* __restrict__ embed_tables,  // (F, V, D)
    const float* __restrict__ att_W,         // (D, A)
    const float* __restrict__ att_b,         // (A,)
    const float* __restrict__ att_h,         // (A,1)
    const float* __restrict__ att_p,         // (D,1)
    const float* __restrict__ lin_w_dense,   // (ND,1)
    const float* __restrict__ lin_w_sparse,  // (F*D,1)
    const float* __restrict__ lin_b,         // (1,)
    float*       __restrict__ out)           // (B,1)
{
    __shared__ __align__(16) float emb_s[ROWS][NF * DD];   // per-row embeddings
    __shared__ float    logit_s[ROWS][NTILE * 16];
    __shared__ unsigned pair_s[NPAIR];          // (i<<8)|j

    const int lane = threadIdx.x & 31;
    const int wv   = threadIdx.x >> 5;
    const int b    = blockIdx.x * ROWS + wv;    // batch row for this wave
    const int n    = lane & 15;                 // column (a / d index)
    const int half = lane >> 4;                 // 0: lanes 0-15, 1: lanes 16-31

    // ---- async gather: embeddings stream straight into LDS (no VGPR staging)
    // 104 chunks of 16B per row; each lane issues scattered global->LDS copies.
    for (int c = lane; c < NF * 4; c += 32) {
        int f = c >> 2;                 // feature (4 chunks per 64B embedding row)
        int q = c & 3;                  // 16B chunk within the embedding row
        int v = sparse_input[b * NF + f];
        const float* g = &embed_tables[((long)f * VOC + v) * DD + q * 4];
        float*       l = &emb_s[wv][c * 4];
#if HAVE_ASYNC_LDS
        __builtin_amdgcn_global_load_async_to_lds_b128(
            (v4i_global_t*)g, (v4i_lds_t*)l, 0, 0);
#else
        *(float4*)l = *(const float4*)g;
#endif
    }

    // ---- cooperative pair table (np.triu_indices(F, k=1) order) ----
    // overlaps with the in-flight async gather
    for (int p = threadIdx.x; p < NPAIR; p += blockDim.x) {
        int i = 0, rem = p;
        while (rem >= NF - 1 - i) { rem -= NF - 1 - i; ++i; }
        pair_s[p] = ((unsigned)i << 8) | (unsigned)(i + 1 + rem);
    }

#if HAVE_ASYNC_LDS
#if __has_builtin(__builtin_amdgcn_s_wait_asynccnt)
    __builtin_amdgcn_s_wait_asynccnt(0);
#else
    asm volatile("s_wait_asynccnt 0x0" ::: "memory");
#endif
#endif
    __syncthreads();   // pair table visible block-wide; emb_s now resident

    // ---- linear term partials (reduced at the end) ----
    float linacc = 0.f;
    for (int e = lane; e < NF * DD; e += 32)
        linacc += emb_s[wv][e] * lin_w_sparse[e];
    for (int e = lane; e < NDNS; e += 32)
        linacc += dense_input[b * NDNS + e] * lin_w_dense[e];

    // ---- B-matrix (att_W, 16x16 f32) resident: four K=4 chunks, 2 VGPRs each
    // 4x16 f32 B layout: VGPR0 -> K = 4c + 2*half, VGPR1 -> K = 4c + 2*half + 1, N = lane&15
    v2f Bc[4];
    #pragma unroll
    for (int c = 0; c < 4; ++c) {
        int k0 = 4 * c + (half << 1);
        Bc[c].x = att_W[k0 * AA + n];
        Bc[c].y = att_W[(k0 + 1) * AA + n];
    }
    const float attb_l = att_b[n];
    const float atth_l = att_h[n];

    // ---- attention MLP via V_WMMA_F32_16X16X4_F32 (K=16 as 4 chained K=4) ----
    for (int t = 0; t < NTILE; ++t) {
        int p     = t * 16 + (lane & 15);       // pair index for A row m
        int valid = (p < NPAIR) ? 1 : 0;
        unsigned pr = pair_s[valid ? p : 0];
        int bi = (int)(pr >> 8) * DD;
        int bj = (int)(pr & 255) * DD;

        v8f acc = {};
        #pragma unroll
        for (int c = 0; c < 4; ++c) {
            // 16x4 f32 A layout: m = lane&15; VGPR0 k = 4c + 2*half, VGPR1 k+1
            int k0 = 4 * c + (half << 1);
            float a0 = emb_s[wv][bi + k0]     * emb_s[wv][bj + k0];
            float a1 = emb_s[wv][bi + k0 + 1] * emb_s[wv][bj + k0 + 1];
            v2f Av;
            Av.x = valid ? a0 : 0.f;
            Av.y = valid ? a1 : 0.f;
            acc = __builtin_amdgcn_wmma_f32_16x16x4_f32(
                false, Av, false, Bc[c], (short)0, acc, false, false);
        }

        // acc[r] = att_wx for row m = r + 8*half, col n.  bias + relu, dot att_h.
        float s[8];
        #pragma unroll
        for (int r = 0; r < 8; ++r) {
            float v = acc[r] + attb_l;
            v = fmaxf(v, 0.f);
            s[r] = v * atth_l;
        }
        // reduce over the 16 columns (lanes within each half)
        #pragma unroll
        for (int m = 1; m <= 8; m <<= 1) {
            #pragma unroll
            for (int r = 0; r < 8; ++r)
                s[r] += __shfl_xor(s[r], m, 32);
        }
        if (n == 0) {
            #pragma unroll
            for (int r = 0; r < 8; ++r) {
                int pm = t * 16 + (half << 3) + r;
                if (pm < NPAIR) logit_s[wv][pm] = s[r];
            }
        }
    }

    // ---- softmax statistics over the 325 logits (wave-local) ----
    float mx = -1e30f;
    for (int p = lane; p < NPAIR; p += 32)
        mx = fmaxf(mx, logit_s[wv][p]);
    #pragma unroll
    for (int m = 1; m <= 16; m <<= 1)
        mx = fmaxf(mx, __shfl_xor(mx, m, 32));

    float sm = 0.f;
    for (int p = lane; p < NPAIR; p += 32)
        sm += expf(logit_s[wv][p] - mx);
    #pragma unroll
    for (int m = 1; m <= 16; m <<= 1)
        sm += __shfl_xor(sm, m, 32);

    // ---- attention-weighted sum of ewp: lane owns dim (lane&15), halves split pairs
    float accd = 0.f;
    for (int p = half; p < NPAIR; p += 2) {
        float w = expf(logit_s[wv][p] - mx);
        unsigned pr = pair_s[p];
        int bi = (int)(pr >> 8) * DD;
        int bj = (int)(pr & 255) * DD;
        accd += w * emb_s[wv][bi + n] * emb_s[wv][bj + n];
    }
    accd += __shfl_xor(accd, 16, 32);           // merge pair halves
    float afm = (accd / sm) * att_p[n];         // project with att_p
    #pragma unroll
    for (int m = 1; m <= 8; m <<= 1)
        afm += __shfl_xor(afm, m, 32);          // sum over the 16 dims

    // ---- reduce linear term over all 32 lanes ----
    #pragma unroll
    for (int m = 1; m <= 16; m <<= 1)
        linacc += __shfl_xor(linacc, m, 32);

    if (lane == 0) {
        float z = afm + linacc + lin_b[0];
        out[b] = 1.f / (1.f + expf(-z));
    }
}

extern "C" void kernel_launch(void* const* d_in, const int* in_sizes, int n_in,
                              void* d_out, int out_size, void* d_ws, size_t ws_size,
                              hipStream_t stream) {
    (void)in_sizes; (void)n_in; (void)d_ws; (void)ws_size; (void)out_size;
    const float* dense_input  = (const float*)d_in[0];
    const int*   sparse_input = (const int*)  d_in[1];
    const float* embed_tables = (const float*)d_in[2];
    const float* att_W        = (const float*)d_in[3];
    const float* att_b        = (const float*)d_in[4];
    const float* att_h        = (const float*)d_in[5];
    const float* att_p        = (const float*)d_in[6];
    const float* lin_w_dense  = (const float*)d_in[7];
    const float* lin_w_sparse = (const float*)d_in[8];
    const float* lin_b        = (const float*)d_in[9];
    float*       out          = (float*)d_out;

    dim3 grid(BATCH / ROWS);
    dim3 block(ROWS * 32);
    afm_kernel<<<grid, block, 0, stream>>>(dense_input, sparse_input, embed_tables,
                                           att_W, att_b, att_h, att_p,
                                           lin_w_dense, lin_w_sparse, lin_b, out);
}